// SketchCritic_17471926960752
// MI455X (gfx1250) — compile-verified
//
#include <hip/hip_runtime.h>
#include <cstdint>

// Problem constants (match reference)
#define B_ 256
#define S_ 512
#define K_ 64
#define ROWS (B_ * S_)

#define GRID_MAIN 4096
#define TPB 256
#define WARPS (TPB / 32)

static constexpr float SIGMA_MIN = 0.01f;
static constexpr float LOG_2PI   = 1.8378770664093453f;

typedef float f2 __attribute__((ext_vector_type(2)));
typedef float f4 __attribute__((ext_vector_type(4)));

// Per-warp LDS staging buffer layout (floats, per 384-float buffer; 1536 B):
//   [  0.. 63]  logits row (64 f)
//   [ 64..191]  mus row   (128 f)
//   [192..255]  sigmas_x  (64 f)
//   [256..319]  sigmas_y  (64 f)
//   [320..383]  sigmas_xy (64 f)
// Two buffers per warp => 768 floats / 3072 B per warp.

__device__ __forceinline__ void async_issue_row(
    const float* __restrict__ logits, const float* __restrict__ mus,
    const float* __restrict__ sx, const float* __restrict__ sy,
    const float* __restrict__ sxy,
    int row, uint32_t lds_buf_byte, int lane)
{
    size_t kb = (size_t)row * K_;
    uint64_t g_lg  = (uint64_t)(uintptr_t)(logits + kb + 2 * lane);
    uint64_t g_mu  = (uint64_t)(uintptr_t)(mus + 2 * kb + 4 * lane);
    uint64_t g_sx  = (uint64_t)(uintptr_t)(sx  + kb + 2 * lane);
    uint64_t g_sy  = (uint64_t)(uintptr_t)(sy  + kb + 2 * lane);
    uint64_t g_sxy = (uint64_t)(uintptr_t)(sxy + kb + 2 * lane);
    uint32_t l_lg  = lds_buf_byte + (uint32_t)lane * 8u;
    uint32_t l_mu  = lds_buf_byte + 256u  + (uint32_t)lane * 16u;
    uint32_t l_sx  = lds_buf_byte + 768u  + (uint32_t)lane * 8u;
    uint32_t l_sy  = lds_buf_byte + 1024u + (uint32_t)lane * 8u;
    uint32_t l_sxy = lds_buf_byte + 1280u + (uint32_t)lane * 8u;
    // CDNA5 async copy engine: global -> LDS, tracked with ASYNCcnt
    asm volatile("global_load_async_to_lds_b64  %0, %1, off" :: "v"(l_lg),  "v"(g_lg)  : "memory");
    asm volatile("global_load_async_to_lds_b128 %0, %1, off" :: "v"(l_mu),  "v"(g_mu)  : "memory");
    asm volatile("global_load_async_to_lds_b64  %0, %1, off" :: "v"(l_sx),  "v"(g_sx)  : "memory");
    asm volatile("global_load_async_to_lds_b64  %0, %1, off" :: "v"(l_sy),  "v"(g_sy)  : "memory");
    asm volatile("global_load_async_to_lds_b64  %0, %1, off" :: "v"(l_sxy), "v"(g_sxy) : "memory");
}

__global__ __launch_bounds__(TPB) void sketch_loss_main(
    const float* __restrict__ xs,        // (B,S,5)
    const float* __restrict__ logits,    // (B,S,K)
    const float* __restrict__ mus,       // (B,S,K,2)
    const float* __restrict__ sxp,       // (B,S,K)
    const float* __restrict__ syp,       // (B,S,K)
    const float* __restrict__ sxyp,      // (B,S,K)
    const float* __restrict__ pen_pred,  // (B,S,3)
    float* __restrict__ partial)         // (GRID_MAIN, 3)
{
    __shared__ __align__(16) float smem[WARPS * 768];
    __shared__ float red[WARPS][3];

    const int lane = threadIdx.x & 31;
    const int warp = threadIdx.x >> 5;
    const int wave = blockIdx.x * WARPS + warp;
    const int nwaves = GRID_MAIN * WARPS;   // 32768 -> 4 rows per wave

    float* wbuf = smem + warp * 768;
    // Low 32 bits of the flat pointer into LDS == wave-relative LDS byte offset.
    const uint32_t wbuf_byte = (uint32_t)(uintptr_t)wbuf;

    float acc_pos = 0.0f, acc_w = 0.0f, acc_wp = 0.0f;

    // Prime the pipeline: stage first row into buffer 0.
    async_issue_row(logits, mus, sxp, syp, sxyp, wave, wbuf_byte, lane);

    int bufsel = 0;
    for (int row = wave; row < ROWS; row += nwaves) {
        const int nrow = row + nwaves;
        if (nrow < ROWS) {
            // Stage next row into the other buffer while we compute this one.
            async_issue_row(logits, mus, sxp, syp, sxyp, nrow,
                            wbuf_byte + (uint32_t)(bufsel ^ 1) * 1536u, lane);
            __builtin_prefetch(xs + (size_t)nrow * 5, 0, 1);        // global_prefetch_b8
            __builtin_prefetch(pen_pred + (size_t)nrow * 3, 0, 1);
            // 5 newer ops outstanding; <=5 means current row's 5 completed (in-order).
            asm volatile("s_wait_asynccnt 5" ::: "memory");
        } else {
            asm volatile("s_wait_asynccnt 0" ::: "memory");
        }

        const float* buf = wbuf + bufsel * 384;
        f2 lg = *(const f2*)(buf + 2 * lane);
        f4 mu = *(const f4*)(buf + 64 + 4 * lane);
        f2 a2 = *(const f2*)(buf + 192 + 2 * lane);
        f2 b2 = *(const f2*)(buf + 256 + 2 * lane);
        f2 c2 = *(const f2*)(buf + 320 + 2 * lane);

        // deltas (wave-uniform): roll(positions,-1) - positions, zeroed at end strokes
        const int s = row & (S_ - 1);
        float dx = 0.0f, dy = 0.0f;
        const float* x0 = xs + (size_t)row * 5;
        if (s != S_ - 1) {
            float p2 = x0[9];                       // xs[b, s+1, 4] (pen_true end flag)
            if (p2 != 1.0f) { dx = x0[5] - x0[0]; dy = x0[6] - x0[1]; }
        }

        // Two mixture components per lane
        float a0 = fmaxf(a2.x, SIGMA_MIN), a1 = fmaxf(a2.y, SIGMA_MIN);
        float b0 = fmaxf(b2.x, SIGMA_MIN), b1 = fmaxf(b2.y, SIGMA_MIN);
        float z00 = (dx - mu.x) / a0;
        float z10 = ((dy - mu.y) - c2.x * z00) / b0;
        float z01 = (dx - mu.z) / a1;
        float z11 = ((dy - mu.w) - c2.y * z01) / b1;
        float comp0 = -0.5f * (z00 * z00 + z10 * z10) - LOG_2PI - logf(a0) - logf(b0);
        float comp1 = -0.5f * (z01 * z01 + z11 * z11) - LOG_2PI - logf(a1) - logf(b1);
        float t0 = lg.x + comp0, t1 = lg.y + comp1;

        // Fused double logsumexp over K=64 via wave32 butterflies
        float m_lg = fmaxf(lg.x, lg.y);
        float m_t  = fmaxf(t0, t1);
        #pragma unroll
        for (int off = 16; off > 0; off >>= 1) {
            m_lg = fmaxf(m_lg, __shfl_xor(m_lg, off, 32));
            m_t  = fmaxf(m_t,  __shfl_xor(m_t,  off, 32));
        }
        float e_lg = expf(lg.x - m_lg) + expf(lg.y - m_lg);
        float e_t  = expf(t0 - m_t)   + expf(t1 - m_t);
        #pragma unroll
        for (int off = 16; off > 0; off >>= 1) {
            e_lg += __shfl_xor(e_lg, off, 32);
            e_t  += __shfl_xor(e_t,  off, 32);
        }
        float log_prob = (m_t + logf(e_t)) - (m_lg + logf(e_lg));
        acc_pos -= log_prob;   // identical in all lanes after reduction

        // Pen loss terms (one-hot targets are exact 0.0/1.0 floats)
        int target = 2;
        if (s != S_ - 1)
            target = (x0[9] == 1.0f) ? 2 : ((x0[8] == 1.0f) ? 1 : 0);
        float w = (target == 1) ? 2.0f : 1.0f;   // PEN_W = {1,2,1}
        float picked = pen_pred[(size_t)row * 3 + target];
        acc_w  += w;
        acc_wp += w * picked;

        bufsel ^= 1;
    }

    if (lane == 0) {
        red[warp][0] = acc_pos; red[warp][1] = acc_w; red[warp][2] = acc_wp;
    }
    __syncthreads();
    if (threadIdx.x == 0) {
        float p = 0.f, w = 0.f, wp = 0.f;
        for (int i = 0; i < WARPS; ++i) { p += red[i][0]; w += red[i][1]; wp += red[i][2]; }
        float* o = partial + (size_t)blockIdx.x * 3;
        o[0] = p; o[1] = w; o[2] = wp;
    }
}

__global__ __launch_bounds__(TPB) void sketch_loss_final(
    const float* __restrict__ partial, float* __restrict__ out)
{
    __shared__ float sp[TPB], sw[TPB], swp[TPB];
    float p = 0.f, w = 0.f, wp = 0.f;
    for (int i = threadIdx.x; i < GRID_MAIN; i += TPB) {
        p  += partial[i * 3 + 0];
        w  += partial[i * 3 + 1];
        wp += partial[i * 3 + 2];
    }
    sp[threadIdx.x] = p; sw[threadIdx.x] = w; swp[threadIdx.x] = wp;
    __syncthreads();
    for (int off = TPB / 2; off > 0; off >>= 1) {
        if ((int)threadIdx.x < off) {
            sp[threadIdx.x]  += sp[threadIdx.x + off];
            sw[threadIdx.x]  += sw[threadIdx.x + off];
            swp[threadIdx.x] += swp[threadIdx.x + off];
        }
        __syncthreads();
    }
    if (threadIdx.x == 0) {
        out[0] = sp[0] / (float)ROWS;   // position_loss = -mean(log_prob)
        out[1] = -swp[0] / sw[0];       // pen_loss
    }
}

extern "C" void kernel_launch(void* const* d_in, const int* in_sizes, int n_in,
                              void* d_out, int out_size, void* d_ws, size_t ws_size,
                              hipStream_t stream)
{
    (void)in_sizes; (void)n_in; (void)out_size; (void)ws_size;
    const float* xs     = (const float*)d_in[0];
    const float* logits = (const float*)d_in[1];
    const float* mus    = (const float*)d_in[2];
    const float* sx     = (const float*)d_in[3];
    const float* sy     = (const float*)d_in[4];
    const float* sxy    = (const float*)d_in[5];
    const float* pen    = (const float*)d_in[6];
    float* partial = (float*)d_ws;   // GRID_MAIN * 3 floats = 48 KB
    float* out     = (float*)d_out;

    sketch_loss_main<<<dim3(GRID_MAIN), dim3(TPB), 0, stream>>>(
        xs, logits, mus, sx, sy, sxy, pen, partial);
    sketch_loss_final<<<dim3(1), dim3(TPB), 0, stream>>>(partial, out);
}